// Alpha_GLM_18021682774403
// MI455X (gfx1250) — compile-verified
//
#include <hip/hip_runtime.h>
#include <hip/hip_bf16.h>

typedef __attribute__((ext_vector_type(2))) float v2f;
typedef __attribute__((ext_vector_type(8))) float v8f;
typedef __attribute__((ext_vector_type(4))) unsigned u32x4;
typedef __attribute__((ext_vector_type(8))) int i32x8;
typedef __attribute__((ext_vector_type(4))) int i32x4;

#define SUB_NO 20
#define T_NO   201

// ---------------------------------------------------------------------------
// Kernel 1: alpha kernels  (sub,2,T_no) summed over the 2 alpha components.
// ---------------------------------------------------------------------------
__global__ void setup_kernel(const float* __restrict__ W_syn,
                             const float* __restrict__ Tau_syn,
                             const float* __restrict__ Delta_syn,
                             float* __restrict__ ekern,
                             float* __restrict__ ikern,
                             float* __restrict__ out_filters) {
  int idx = blockIdx.x * blockDim.x + threadIdx.x;
  const int total = SUB_NO * 2 * T_NO;
  if (idx >= total) return;
  int t = idx % T_NO;
  int rest = idx / T_NO;
  int c = rest % 2;       // 0 = excitatory, 1 = inhibitory
  int s = rest / 2;
  float sum = 0.f;
#pragma unroll
  for (int j = 0; j < 2; ++j) {
    float w   = W_syn[(s * 2 + j) * 2 + c];
    float tau = Tau_syn[(s * 2 + j) * 2 + c];
    float d   = Delta_syn[(s * 2 + j) * 2 + c];
    float td  = fmaxf((float)t - d, 0.f);
    float tt  = td * __expf(-tau);            // td / exp(tau)
    sum += tt * __expf(-tt) * w;
  }
  float* dst = c ? ikern : ekern;
  dst[s * T_NO + t] = sum;
  out_filters[(c * SUB_NO + s) * T_NO + t] = sum;   // rows 0..19 = e, 20..39 = i
}

// ---------------------------------------------------------------------------
// Kernel 2: syn aggregation GEMM via V_WMMA_F32_16X16X4_F32 with TDM
// (tensor_load_to_lds) staging of the A tile.
//
// A tile: 128 rows x 40 cols f32, DMA'd by the Tensor Data Mover with
// pad_enable (1 DWORD pad per 8 DWORDs) -> effective LDS row stride 45
// DWORDs, gcd(45,64)=1 -> conflict-free WMMA A reads.
// LDS dword index of element (r,k):  r*45 + k + (k>>3)
// OOB rows (row >= T) read zero via tensor_dim1 = T - row0.
// ---------------------------------------------------------------------------
#define GROWS 128
#define KCH   40
#define ASTRIDE 45   // 40 + 40/8 pad dwords

__device__ __forceinline__ int a_idx(int r, int k) {
  return r * ASTRIDE + k + (k >> 3);
}

__global__ __launch_bounds__(256) void gemm_kernel(
    const float* __restrict__ S_e, const float* __restrict__ S_i,
    const float* __restrict__ Ce,  const float* __restrict__ Ci,
    float* __restrict__ syn_e, float* __restrict__ syn_i, int T) {
  __shared__ float Alds[GROWS * ASTRIDE];
  __shared__ float Blds[KCH * 33];

  const int tid  = threadIdx.x;
  const int lane = tid & 31;
  const int wave = tid >> 5;
  const int row0 = blockIdx.x * GROWS;

  const int m    = lane & 15;          // A row within tile / B column
  const int koff = (lane >> 4) << 1;   // K sub-offset (ISA 16x4 f32 A layout)
  const int wr   = wave * 16;

  for (int phase = 0; phase < 2; ++phase) {
    const float* __restrict__ S = phase ? S_i : S_e;
    const float* __restrict__ C = phase ? Ci : Ce;
    float* __restrict__ O       = phase ? syn_i : syn_e;
    const int K                 = phase ? 200 : 800;

    v8f c0 = {};
    v8f c1 = {};

    for (int kb = 0; kb < K; kb += KCH) {
      __syncthreads();   // previous chunk's compute finished before overwrite

#if __has_builtin(__builtin_amdgcn_tensor_load_to_lds)
      // --- TDM: DMA the 128x40 A tile global -> LDS (one issue per block) ---
      if (wave == 0) {
        unsigned lds0 = (unsigned)(uintptr_t)&Alds[0];
        unsigned long long ga =
            (unsigned long long)(uintptr_t)S +
            (((unsigned long long)(unsigned)row0 * (unsigned)K + (unsigned)kb) << 2);
        u32x4 g0;
        g0[0] = 1u;                                    // count=1, user descriptor
        g0[1] = lds0;                                  // lds_addr
        g0[2] = (unsigned)ga;                          // global_addr[31:0]
        g0[3] = (unsigned)((ga >> 32) & 0x1FFFFFFu) |  // global_addr[56:32]
                (2u << 30);                            // type = 2 ("image")
        int dim0 = K - kb;                             // remaining cols
        int dim1 = T - row0;                           // remaining rows (OOB->0)
        i32x8 g1;
        g1[0] = (2 << 16)        // data_size = 4 bytes
              | (1 << 20)        // pad_enable
              | (2 << 22);       // pad_interval: 8 dwords (pad_amount code 0 = 1 dword)
        g1[1] = (dim0 & 0xFFFF) << 16;                       // tensor_dim0[15:0]
        g1[2] = ((dim1 & 0xFFFF) << 16) | ((dim0 >> 16) & 0xFFFF);
        g1[3] = (KCH << 16) | ((dim1 >> 16) & 0xFFFF);       // tile_dim0 | dim1 hi
        g1[4] = GROWS;                                       // tile_dim1 (tile_dim2=0)
        g1[5] = K;                                           // tensor_dim0_stride lo32
        g1[6] = 0;
        g1[7] = 0;
        i32x4 gz  = {0, 0, 0, 0};                            // 2D: groups 2/3 unused
        i32x8 gz8 = {0, 0, 0, 0, 0, 0, 0, 0};                // unused extra group
        __builtin_amdgcn_tensor_load_to_lds(g0, g1, gz, gz, gz8, 0);
      }
#else
      // --- fallback: synchronous staging, same padded layout ---
      for (int i = tid; i < GROWS * KCH; i += 256) {
        int r = i / KCH, k = i % KCH;
        int row = row0 + r;
        Alds[a_idx(r, k)] = (row < T) ? S[(size_t)row * K + kb + k] : 0.f;
      }
#endif

      // Stage B: 40 x 32 ([k][n], stride 33; cols >= 20 zero padding).
      for (int i = tid; i < KCH * 32; i += 256) {
        int k = i >> 5, n = i & 31;
        Blds[k * 33 + n] = (n < SUB_NO) ? C[n * K + kb + k] : 0.f;
      }

#if __has_builtin(__builtin_amdgcn_tensor_load_to_lds)
      if (wave == 0) __builtin_amdgcn_s_wait_tensorcnt(0);
#endif
      __syncthreads();   // A (TDM) + B visible to all waves

#pragma unroll
      for (int kk = 0; kk < KCH; kk += 4) {
        v2f a, b0, b1;
        const int kc = kk + koff;
        a.x  = Alds[a_idx(wr + m, kc)];
        a.y  = Alds[a_idx(wr + m, kc + 1)];
        b0.x = Blds[kc * 33 + m];
        b0.y = Blds[(kc + 1) * 33 + m];
        b1.x = Blds[kc * 33 + 16 + m];
        b1.y = Blds[(kc + 1) * 33 + 16 + m];
        c0 = __builtin_amdgcn_wmma_f32_16x16x4_f32(false, a, false, b0,
                                                   (short)0, c0, false, false);
        c1 = __builtin_amdgcn_wmma_f32_16x16x4_f32(false, a, false, b1,
                                                   (short)0, c1, false, false);
      }
    }

    // C/D layout: VGPR v holds M = v + 8*(lane>>4), N = lane&15.
    const int n     = lane & 15;
    const int rbase = wr + ((lane >> 4) << 3);
#pragma unroll
    for (int v = 0; v < 8; ++v) {
      int row = row0 + rbase + v;
      if (row < T) {
        O[(size_t)row * SUB_NO + n] = c0[v];
        if (n + 16 < SUB_NO) O[(size_t)row * SUB_NO + n + 16] = c1[v];
      }
    }
    __syncthreads();
  }
}

// ---------------------------------------------------------------------------
// Kernel 3: fused causal grouped conv (e + i) + dendritic tree + tanh + volt.
// 640 threads = 20 waves; wave s owns subunit s; each lane computes 4
// consecutive t via a sliding 4-register window (full unroll -> zero moves).
// ---------------------------------------------------------------------------
#define CCT 128   // t per block

__global__ __launch_bounds__(640) void conv_tree_kernel(
    const float* __restrict__ syn_e, const float* __restrict__ syn_i,
    const float* __restrict__ ekern, const float* __restrict__ ikern,
    const float* __restrict__ C_den, const float* __restrict__ W_sub,
    const float* __restrict__ Theta, const float* __restrict__ V_o,
    float* __restrict__ volt, int T) {
  __shared__ float xe[SUB_NO][CCT + T_NO - 1];   // 20 x 328
  __shared__ float xi[SUB_NO][CCT + T_NO - 1];
  __shared__ float sin_lds[CCT][SUB_NO + 1];     // syn_in tile
  __shared__ float We[SUB_NO];

  const int tid  = threadIdx.x;
  const int lane = tid & 31;
  const int s    = tid >> 5;            // subunit = wave id (20 waves)
  const int t0   = blockIdx.x * CCT;
  const int HALO = T_NO - 1;            // 200

  if (tid < SUB_NO) We[tid] = __expf(W_sub[tid]);

  // Stage signal tiles: global t in [t0-200, t0+CCT), coalesced ((t,s) linear).
  for (int i = tid; i < SUB_NO * (CCT + HALO); i += 640) {
    int j = i / SUB_NO, ss = i % SUB_NO;
    int t = t0 - HALO + j;
    float ve = 0.f, vi = 0.f;
    if (t >= 0 && t < T) {
      ve = syn_e[(size_t)t * SUB_NO + ss];
      vi = syn_i[(size_t)t * SUB_NO + ss];
    }
    xe[ss][j] = ve;
    xi[ss][j] = vi;
  }
  __syncthreads();

  // Conv: each lane computes 4 consecutive outputs for subunit s.
  {
    const int j0 = HALO + lane * 4;     // LDS index of first output time
    float a0 = 0.f, a1 = 0.f, a2 = 0.f, a3 = 0.f;

    float w0 = xe[s][j0], w1 = xe[s][j0 + 1], w2 = xe[s][j0 + 2], w3 = xe[s][j0 + 3];
#pragma unroll
    for (int k = 0; k < T_NO; ++k) {
      float kv = ekern[s * T_NO + k];   // wave-uniform, L2-broadcast
      a0 += kv * w0; a1 += kv * w1; a2 += kv * w2; a3 += kv * w3;
      w3 = w2; w2 = w1; w1 = w0;
      w0 = xe[s][j0 - 1 - k];
    }
    w0 = xi[s][j0]; w1 = xi[s][j0 + 1]; w2 = xi[s][j0 + 2]; w3 = xi[s][j0 + 3];
#pragma unroll
    for (int k = 0; k < T_NO; ++k) {
      float kv = ikern[s * T_NO + k];
      a0 += kv * w0; a1 += kv * w1; a2 += kv * w2; a3 += kv * w3;
      w3 = w2; w2 = w1; w1 = w0;
      w0 = xi[s][j0 - 1 - k];
    }
    sin_lds[lane * 4 + 0][s] = a0;
    sin_lds[lane * 4 + 1][s] = a1;
    sin_lds[lane * 4 + 2][s] = a2;
    sin_lds[lane * 4 + 3][s] = a3;
  }
  __syncthreads();

  // Tree recursion, one thread per t. C_den is strictly upper triangular
  // (children indices > parent), so only c > s2 contributes.
  if (tid < CCT) {
    int t = t0 + tid;
    if (t < T) {
      float so[SUB_NO];
#pragma unroll
      for (int s2 = SUB_NO - 1; s2 >= 0; --s2) {
        float ls = 0.f;
#pragma unroll
        for (int c = SUB_NO - 1; c > s2; --c)
          ls += C_den[s2 * SUB_NO + c] * We[c] * so[c];
        so[s2] = tanhf(sin_lds[tid][s2] + ls + Theta[s2]);
      }
      volt[t] = so[0] * We[0] + V_o[0];
    }
  }
}

// ---------------------------------------------------------------------------
// Launch
// ---------------------------------------------------------------------------
extern "C" void kernel_launch(void* const* d_in, const int* in_sizes, int n_in,
                              void* d_out, int out_size, void* d_ws, size_t ws_size,
                              hipStream_t stream) {
  const float* S_e     = (const float*)d_in[0];
  const float* S_i     = (const float*)d_in[1];
  const float* C_syn_e = (const float*)d_in[2];
  const float* C_syn_i = (const float*)d_in[3];
  const float* C_den   = (const float*)d_in[4];
  const float* W_syn   = (const float*)d_in[5];
  const float* Tau_syn = (const float*)d_in[6];
  const float* Delta   = (const float*)d_in[7];
  const float* W_sub   = (const float*)d_in[8];
  const float* Theta   = (const float*)d_in[9];
  const float* V_o     = (const float*)d_in[10];

  const int E_NO = 800, I_NO = 200;
  const int T = in_sizes[0] / E_NO;     // 100000

  // Workspace layout (floats)
  float* ws    = (float*)d_ws;
  float* syn_e = ws;                                   // T*20
  float* syn_i = ws + (size_t)T * SUB_NO;              // T*20
  float* ekern = ws + (size_t)2 * T * SUB_NO;          // 20*201
  float* ikern = ekern + SUB_NO * T_NO;                // 20*201

  // d_out layout (floats): volt[T] | out_filters[40*201] | C_syn_e | C_syn_i
  float* out         = (float*)d_out;
  float* volt        = out;
  float* out_filters = out + T;
  float* out_ce      = out_filters + 2 * SUB_NO * T_NO;
  float* out_ci      = out_ce + SUB_NO * E_NO;

  // 1. alpha kernels
  setup_kernel<<<(SUB_NO * 2 * T_NO + 255) / 256, 256, 0, stream>>>(
      W_syn, Tau_syn, Delta, ekern, ikern, out_filters);

  // 2. pass-through outputs (device-to-device, graph-capture safe)
  (void)hipMemcpyAsync(out_ce, C_syn_e, (size_t)SUB_NO * E_NO * sizeof(float),
                       hipMemcpyDeviceToDevice, stream);
  (void)hipMemcpyAsync(out_ci, C_syn_i, (size_t)SUB_NO * I_NO * sizeof(float),
                       hipMemcpyDeviceToDevice, stream);

  // 3. WMMA aggregation GEMMs (TDM-staged A tiles)
  gemm_kernel<<<(T + GROWS - 1) / GROWS, 256, 0, stream>>>(
      S_e, S_i, C_syn_e, C_syn_i, syn_e, syn_i, T);

  // 4. fused conv + tree + voltage
  conv_tree_kernel<<<(T + CCT - 1) / CCT, 640, 0, stream>>>(
      syn_e, syn_i, ekern, ikern, C_den, W_sub, Theta, V_o, volt, T);
}